// MultiHeadGATLayer_2654289789413
// MI455X (gfx1250) — compile-verified
//
#include <hip/hip_runtime.h>
#include <math.h>

#define N_NODES 30000
#define E_EDGES 480000
#define IN_DIMC 512
#define OUT_DIMC 64
#define H_HEADS 8
#define D_HIDC 2048
#define LN_EPSF 1e-5f
#define NEG_SLOPEF 0.01f

typedef float v2f __attribute__((ext_vector_type(2)));
typedef float v8f __attribute__((ext_vector_type(8)));

// ---------------------------------------------------------------------------
// fp32 WMMA GEMM:  C[n,m] = act( sum_k A[n,k]*W[m,k] + bias[m] ) (+ res[n,m])
// A: (Nrows x K) row-major;  W: (Mcols x K) row-major (i.e. we compute A@W^T)
// grid = (Mcols/64, ceil(Nrows/128)), block = 256 (8 wave32)
// Block tile 128x64; wave (wr,wc) owns 4 stacked 16x16 tiles at rows
// wr*64 + t*16, cols wc*16.  Double-buffered LDS: one barrier per K-chunk.
// ---------------------------------------------------------------------------
template<int ACT> // 0 = none, 1 = relu
__global__ __launch_bounds__(256)
void gemm_wmma_f32(const float* __restrict__ A, const float* __restrict__ W,
                   const float* __restrict__ bias, const float* __restrict__ res,
                   float* __restrict__ C, int Nrows, int Mcols, int K)
{
    __shared__ float As[2][128][20];  // stride 20 floats: 16B-aligned, conflict-free
    __shared__ float Ws[2][64][20];

    const int tid  = threadIdx.x;
    const int w    = tid >> 5;       // wave 0..7
    const int lane = tid & 31;
    const int wr   = w >> 2;         // 0..1  (row group of 64)
    const int wc   = w & 3;          // 0..3  (col group of 16)
    const int half = lane >> 4;      // 0/1
    const int l15  = lane & 15;

    const int rowBlock = blockIdx.y * 128;
    const int colBlock = blockIdx.x * 64;

    v8f acc[4];
    #pragma unroll
    for (int t = 0; t < 4; ++t)
        acc[t] = (v8f){0.f,0.f,0.f,0.f,0.f,0.f,0.f,0.f};

    const int ldRow = tid >> 2;        // 0..63
    const int ldK   = (tid & 3) * 4;   // 0,4,8,12

    auto stage = [&](int buf, int kc) {
        // A tile: 128 rows x 16 K  (two float4 per thread)
        {
            const int gr = rowBlock + ldRow;
            float4 v = make_float4(0.f, 0.f, 0.f, 0.f);
            if (gr < Nrows)
                v = *(const float4*)(A + (size_t)gr * K + kc + ldK);
            As[buf][ldRow][ldK + 0] = v.x; As[buf][ldRow][ldK + 1] = v.y;
            As[buf][ldRow][ldK + 2] = v.z; As[buf][ldRow][ldK + 3] = v.w;

            const int gr2 = rowBlock + 64 + ldRow;
            float4 v2 = make_float4(0.f, 0.f, 0.f, 0.f);
            if (gr2 < Nrows)
                v2 = *(const float4*)(A + (size_t)gr2 * K + kc + ldK);
            As[buf][64 + ldRow][ldK + 0] = v2.x; As[buf][64 + ldRow][ldK + 1] = v2.y;
            As[buf][64 + ldRow][ldK + 2] = v2.z; As[buf][64 + ldRow][ldK + 3] = v2.w;
        }
        // W tile: 64 out-features x 16 K; Mcols % 64 == 0
        {
            const int gm = colBlock + ldRow;
            float4 v = *(const float4*)(W + (size_t)gm * K + kc + ldK);
            Ws[buf][ldRow][ldK + 0] = v.x; Ws[buf][ldRow][ldK + 1] = v.y;
            Ws[buf][ldRow][ldK + 2] = v.z; Ws[buf][ldRow][ldK + 3] = v.w;
        }
    };

    const int nch = K >> 4;
    stage(0, 0);
    __syncthreads();

    for (int c = 0; c < nch; ++c) {
        const int cur = c & 1;
        if (c + 1 < nch)
            stage(cur ^ 1, (c + 1) << 4);  // overlaps with compute on `cur`

        #pragma unroll
        for (int k0 = 0; k0 < 16; k0 += 4) {
            const int kk = k0 + 2 * half;  // lanes 0-15: K=k0,k0+1 ; lanes 16-31: K=k0+2,k0+3
            v2f b;
            b.x = Ws[cur][wc * 16 + l15][kk];
            b.y = Ws[cur][wc * 16 + l15][kk + 1];
            #pragma unroll
            for (int t = 0; t < 4; ++t) {
                v2f a;
                a.x = As[cur][wr * 64 + t * 16 + l15][kk];
                a.y = As[cur][wr * 64 + t * 16 + l15][kk + 1];
                acc[t] = __builtin_amdgcn_wmma_f32_16x16x4_f32(false, a, false, b,
                                                               (short)0, acc[t], false, false);
            }
        }
        __syncthreads();
    }

    // ---- epilogue: C/D layout: VGPR v -> row v (+8 for upper lanes), l15 -> col ----
    const int col = colBlock + wc * 16 + l15;
    const float bv = bias ? bias[col] : 0.f;
    #pragma unroll
    for (int t = 0; t < 4; ++t) {
        #pragma unroll
        for (int v = 0; v < 8; ++v) {
            const int row = rowBlock + wr * 64 + t * 16 + v + 8 * half;
            if (row < Nrows) {
                float x = acc[t][v] + bv;
                if (ACT == 1) x = x > 0.f ? x : 0.f;
                if (res) x += res[(size_t)row * Mcols + col];
                C[(size_t)row * Mcols + col] = x;
            }
        }
    }
}

// ---------------------------------------------------------------------------
// attention scalar scores: s_src[n,h] = z[n,h,:].a_src[h], s_dst analog
// ---------------------------------------------------------------------------
__global__ __launch_bounds__(256)
void scores_kernel(const float* __restrict__ z, const float* __restrict__ Wa,
                   float* __restrict__ s_src, float* __restrict__ s_dst)
{
    const int i = blockIdx.x * blockDim.x + threadIdx.x;  // n*H + h
    if (i >= N_NODES * H_HEADS) return;
    const int n = i >> 3, h = i & 7;
    const float* zr = z + (size_t)n * IN_DIMC + h * OUT_DIMC;
    const float* wa = Wa + h * 2 * OUT_DIMC;
    float ss = 0.f, sd = 0.f;
    #pragma unroll 8
    for (int o = 0; o < OUT_DIMC; ++o) {
        const float zv = zr[o];
        ss += zv * wa[o];
        sd += zv * wa[OUT_DIMC + o];
    }
    s_src[i] = ss;
    s_dst[i] = sd;
}

// order-preserving float<->uint encoding for atomic max over signed floats
__device__ __forceinline__ unsigned f2ord(float f) {
    unsigned u = __float_as_uint(f);
    return (u & 0x80000000u) ? ~u : (u | 0x80000000u);
}
__device__ __forceinline__ float ord2f(unsigned u) {
    return __uint_as_float((u & 0x80000000u) ? (u & 0x7FFFFFFFu) : ~u);
}

__global__ __launch_bounds__(256)
void edge_max_kernel(const int* __restrict__ src, const int* __restrict__ dst,
                     const float* __restrict__ s_src, const float* __restrict__ s_dst,
                     unsigned* __restrict__ menc)
{
    const int i = blockIdx.x * blockDim.x + threadIdx.x;  // e*H + h
    if (i >= E_EDGES * H_HEADS) return;
    const int e = i >> 3, h = i & 7;
    const float s = s_src[src[e] * H_HEADS + h] + s_dst[dst[e] * H_HEADS + h];
    const float ls = s > 0.f ? s : NEG_SLOPEF * s;
    atomicMax(&menc[dst[e] * H_HEADS + h], f2ord(ls));
}

__global__ __launch_bounds__(256)
void edge_exp_kernel(const int* __restrict__ src, const int* __restrict__ dst,
                     const float* __restrict__ s_src, const float* __restrict__ s_dst,
                     const unsigned* __restrict__ menc,
                     float* __restrict__ exbuf, float* __restrict__ denom)
{
    const int i = blockIdx.x * blockDim.x + threadIdx.x;  // e*H + h
    if (i >= E_EDGES * H_HEADS) return;
    const int e = i >> 3, h = i & 7;
    const int d = dst[e];
    const float s = s_src[src[e] * H_HEADS + h] + s_dst[d * H_HEADS + h];
    const float ls = s > 0.f ? s : NEG_SLOPEF * s;
    const float ex = expf(ls - ord2f(menc[d * H_HEADS + h]));
    exbuf[i] = ex;
    atomicAdd(&denom[d * H_HEADS + h], ex);
}

// scatter-add messages: gat[dst, h*64+o] += alpha * z[src, h*64+o]; 4 floats/thread
__global__ __launch_bounds__(256)
void edge_agg_kernel(const int* __restrict__ src, const int* __restrict__ dst,
                     const float* __restrict__ exbuf, const float* __restrict__ denom,
                     const float* __restrict__ z, float* __restrict__ gat)
{
    const long long i = (long long)blockIdx.x * blockDim.x + threadIdx.x;  // e*128 + r
    if (i >= (long long)E_EDGES * 128) return;
    const int e = (int)(i >> 7);
    const int r = (int)(i & 127);
    const int h = r >> 4;
    const int o = (r & 15) * 4;
    const int sN = src[e], dN = dst[e];
    const float alpha = exbuf[e * H_HEADS + h] / denom[dN * H_HEADS + h];
    const float4 zv = *(const float4*)(z + (size_t)sN * IN_DIMC + h * OUT_DIMC + o);
    float* out = gat + (size_t)dN * IN_DIMC + h * OUT_DIMC + o;
    atomicAdd(out + 0, alpha * zv.x);
    atomicAdd(out + 1, alpha * zv.y);
    atomicAdd(out + 2, alpha * zv.z);
    atomicAdd(out + 3, alpha * zv.w);
}

// x = elu(gat) + h   (in-place on gat buffer)
__global__ __launch_bounds__(256)
void elu_residual_kernel(float* __restrict__ gat, const float* __restrict__ h)
{
    const int i = blockIdx.x * blockDim.x + threadIdx.x;
    if (i >= N_NODES * IN_DIMC) return;
    const float v = gat[i];
    const float e = v > 0.f ? v : (expf(v) - 1.f);
    gat[i] = e + h[i];
}

// one 256-thread block per row of 512; mean/var reduce, write normalized output
__global__ __launch_bounds__(256)
void layernorm_kernel(const float* __restrict__ y, const float* __restrict__ g,
                      const float* __restrict__ b, float* __restrict__ out)
{
    __shared__ float red[256];
    const int n = blockIdx.x;
    const int t = threadIdx.x;
    const float* yr = y + (size_t)n * IN_DIMC;
    const float v0 = yr[t], v1 = yr[t + 256];

    red[t] = v0 + v1;
    __syncthreads();
    for (int st = 128; st > 0; st >>= 1) {
        if (t < st) red[t] += red[t + st];
        __syncthreads();
    }
    const float mu = red[0] * (1.f / IN_DIMC);
    __syncthreads();

    const float d0 = v0 - mu, d1 = v1 - mu;
    red[t] = d0 * d0 + d1 * d1;
    __syncthreads();
    for (int st = 128; st > 0; st >>= 1) {
        if (t < st) red[t] += red[t + st];
        __syncthreads();
    }
    const float inv = rsqrtf(red[0] * (1.f / IN_DIMC) + LN_EPSF);

    float* outr = out + (size_t)n * IN_DIMC;
    outr[t]       = d0 * inv * g[t]       + b[t];
    outr[t + 256] = d1 * inv * g[t + 256] + b[t + 256];
}

// ---------------------------------------------------------------------------
extern "C" void kernel_launch(void* const* d_in, const int* in_sizes, int n_in,
                              void* d_out, int out_size, void* d_ws, size_t ws_size,
                              hipStream_t stream)
{
    const float* h    = (const float*)d_in[0];
    const int*   src  = (const int*)  d_in[1];
    const int*   dst  = (const int*)  d_in[2];
    const float* Wfc  = (const float*)d_in[3];   // (8,64,512) == (512,512) row-major
    const float* Wa   = (const float*)d_in[4];   // (8,128)
    const float* w1   = (const float*)d_in[5];   // (2048,512)
    const float* b1   = (const float*)d_in[6];
    const float* w2   = (const float*)d_in[7];   // (512,2048)
    const float* b2   = (const float*)d_in[8];
    const float* ln_g = (const float*)d_in[9];
    const float* ln_b = (const float*)d_in[10];
    float* out = (float*)d_out;

    // workspace layout (floats)
    float*    z     = (float*)d_ws;                       // 30000*512
    float*    xbuf  = z     + (size_t)N_NODES * IN_DIMC;  // 30000*512 (gat, then x)
    float*    s_src = xbuf  + (size_t)N_NODES * IN_DIMC;  // 30000*8
    float*    s_dst = s_src + (size_t)N_NODES * H_HEADS;  // 30000*8
    unsigned* menc  = (unsigned*)(s_dst + (size_t)N_NODES * H_HEADS); // 30000*8
    float*    denom = (float*)(menc + (size_t)N_NODES * H_HEADS);     // 30000*8
    float*    exbuf = denom + (size_t)N_NODES * H_HEADS;  // 480000*8
    float*    y1    = exbuf + (size_t)E_EDGES * H_HEADS;  // 30000*2048

    const int rowBlocks = (N_NODES + 127) / 128;          // 235

    // 1) z = h @ Wfc^T        (30000x512 @ 512x512)
    gemm_wmma_f32<0><<<dim3(IN_DIMC / 64, rowBlocks), 256, 0, stream>>>(
        h, Wfc, nullptr, nullptr, z, N_NODES, IN_DIMC, IN_DIMC);

    // 2) attention scalar scores
    scores_kernel<<<(N_NODES * H_HEADS + 255) / 256, 256, 0, stream>>>(z, Wa, s_src, s_dst);

    // 3) init segment-max (ordered-uint 0 == below all reals), denom, gat accumulator
    hipMemsetAsync(menc, 0, (size_t)N_NODES * H_HEADS * sizeof(unsigned), stream);
    hipMemsetAsync(denom, 0, (size_t)N_NODES * H_HEADS * sizeof(float), stream);
    hipMemsetAsync(xbuf, 0, (size_t)N_NODES * IN_DIMC * sizeof(float), stream);

    // 4) segment max over dst
    edge_max_kernel<<<(E_EDGES * H_HEADS + 255) / 256, 256, 0, stream>>>(
        src, dst, s_src, s_dst, menc);

    // 5) exp + segment sum
    edge_exp_kernel<<<(E_EDGES * H_HEADS + 255) / 256, 256, 0, stream>>>(
        src, dst, s_src, s_dst, menc, exbuf, denom);

    // 6) weighted scatter-add of messages
    {
        const long long work = (long long)E_EDGES * 128;
        edge_agg_kernel<<<(unsigned)((work + 255) / 256), 256, 0, stream>>>(
            src, dst, exbuf, denom, z, xbuf);
    }

    // 7) x = elu(gat) + h
    elu_residual_kernel<<<(N_NODES * IN_DIMC + 255) / 256, 256, 0, stream>>>(xbuf, h);

    // 8) y1 = relu(x @ w1^T + b1)      (30000x512 @ 512x2048)
    gemm_wmma_f32<1><<<dim3(D_HIDC / 64, rowBlocks), 256, 0, stream>>>(
        xbuf, w1, b1, nullptr, y1, N_NODES, D_HIDC, IN_DIMC);

    // 9) y = y1 @ w2^T + b2 + x        (reuse z region for pre-LN result)
    gemm_wmma_f32<0><<<dim3(IN_DIMC / 64, rowBlocks), 256, 0, stream>>>(
        y1, w2, b2, xbuf, z, N_NODES, IN_DIMC, D_HIDC);

    // 10) layernorm -> d_out
    layernorm_kernel<<<N_NODES, 256, 0, stream>>>(z, ln_g, ln_b, out);
}